// TNEP_67070209294798
// MI455X (gfx1250) — compile-verified
//
#include <hip/hip_runtime.h>
#include <hip/hip_bf16.h>
#include <math.h>

// Problem constants (match reference)
#define NB    16
#define NA    512
#define NQ    64
#define NM    4000
#define NATOM (NB * NA)        // 8192
#define NWAVES 8               // waves per block, each takes m-tiles strided
#define MSTEP  (NWAVES * 16)   // m stride per wave (128)

typedef __attribute__((ext_vector_type(2))) float v2f;
typedef __attribute__((ext_vector_type(8))) float v8f;

// ---------------------------------------------------------------------------
// Prep 1: per-atom inverse norm  rnorm[i] = 1 / max(||desc_i||, 1e-12)
// ---------------------------------------------------------------------------
__global__ __launch_bounds__(256) void tnep_rnorm_kernel(
    const float* __restrict__ desc, float* __restrict__ rnorm)
{
    const int wave = (blockIdx.x * blockDim.x + threadIdx.x) >> 5;  // atom id
    const int lane = threadIdx.x & 31;
    if (wave >= NATOM) return;
    const float* d = desc + (size_t)wave * NQ;
    float x0 = d[lane];
    float x1 = d[lane + 32];
    float ss = x0 * x0 + x1 * x1;
    #pragma unroll
    for (int off = 16; off > 0; off >>= 1)
        ss += __shfl_xor(ss, off, 32);
    if (lane == 0)
        rnorm[wave] = 1.0f / fmaxf(sqrtf(ss), 1e-12f);
}

// ---------------------------------------------------------------------------
// Prep 2: per-sparse-point coefficient  coef[m] = alpha[m] * delta[spec]^2
// ---------------------------------------------------------------------------
__global__ __launch_bounds__(256) void tnep_coef_kernel(
    const int* __restrict__ species, const float* __restrict__ alpha,
    const float* __restrict__ delta, float* __restrict__ coef)
{
    const int i = blockIdx.x * blockDim.x + threadIdx.x;
    if (i >= NM) return;
    int s = species[i];
    if (s < 0) s = 0;
    const float dlt = delta[s];
    coef[i] = alpha[i] * dlt * dlt;
}

// ---------------------------------------------------------------------------
// Main kernel: 8 waves per block; block owns a 16-atom tile; waves split the
// M range (stride 128). Per m-tile: 16x V_WMMA_F32_16X16X4_F32 over K=64,
// double-buffered B loads, two independent accumulation chains.
//
// Fragment layouts (wave32, per CDNA5 ISA 7.12.2):
//   A 16x4 f32 : lane -> row M=lane&15 ; VGPR0/1 -> K = 2*half, 2*half+1
//   B 4x16 f32 : lane -> col N=lane&15 ; VGPR0/1 -> K = 2*half, 2*half+1
//   C/D 16x16  : VGPR v -> row M = v + 8*half ; lane -> col N = lane&15
// ---------------------------------------------------------------------------
__global__ __launch_bounds__(32 * NWAVES) void tnep_u_kernel(
    const float* __restrict__ desc,       // [NATOM, NQ]
    const int*   __restrict__ Z,          // [NATOM]
    const float* __restrict__ amask,      // [NATOM]
    const float* __restrict__ sq,         // [NM, NQ]
    const int*   __restrict__ species,    // [NM]
    const float* __restrict__ rnorm,      // [NATOM]
    const float* __restrict__ coef,       // [NM]
    float*       __restrict__ U)          // [NATOM]
{
    const int atom0 = blockIdx.x * 16;      // base atom row of this tile
    const int tid   = threadIdx.x;
    const int wave  = tid >> 5;             // 0..NWAVES-1
    const int lane  = tid & 31;
    const int half  = lane >> 4;            // 0 or 1
    const int l16   = lane & 15;
    const int kbase = half * 2;             // K offset of this half-wave

    __shared__ float red[NWAVES][16];

    // --- Load the 16x64 atom tile once: 16 k-step fragments per lane -------
    const float* arow = desc + (size_t)(atom0 + l16) * NQ + kbase;
    v2f a[16];
    #pragma unroll
    for (int kk = 0; kk < 16; ++kk) {
        a[kk].x = arow[4 * kk + 0];
        a[kk].y = arow[4 * kk + 1];
    }

    // Per-accumulator-VGPR row metadata (row M = v + 8*half)
    int   zv[8];
    float rn[8];
    #pragma unroll
    for (int v = 0; v < 8; ++v) {
        const int row = atom0 + v + half * 8;
        zv[v] = Z[row];
        rn[v] = rnorm[row];
    }

    float uacc[8];
    #pragma unroll
    for (int v = 0; v < 8; ++v) uacc[v] = 0.0f;

    // --- Prologue: load first m-tile for this wave --------------------------
    int m0 = wave * 16;                      // always < NM (NWAVES*16 <= NM)
    v2f   b[16];
    int   spec;
    float cf;
    {
        const float* brow = sq + (size_t)(m0 + l16) * NQ + kbase;
        #pragma unroll
        for (int kk = 0; kk < 16; ++kk) {
            b[kk].x = brow[4 * kk + 0];
            b[kk].y = brow[4 * kk + 1];
        }
        spec = species[m0 + l16];
        cf   = coef[m0 + l16];
    }

    // --- Main loop: prefetch next tile, compute current ---------------------
    while (m0 < NM) {
        const int m1 = m0 + MSTEP;
        v2f   b2[16];
        int   spec2 = 0;
        float cf2   = 0.0f;
        if (m1 < NM) {                       // wave-uniform condition
            const float* brow2 = sq + (size_t)(m1 + l16) * NQ + kbase;
            #pragma unroll
            for (int kk = 0; kk < 16; ++kk) {
                b2[kk].x = brow2[4 * kk + 0];
                b2[kk].y = brow2[4 * kk + 1];
            }
            spec2 = species[m1 + l16];
            cf2   = coef[m1 + l16];
        }

        // Two independent accumulation chains over K=64
        v8f acc0 = {};
        v8f acc1 = {};
        #pragma unroll
        for (int kk = 0; kk < 16; kk += 2) {
            acc0 = __builtin_amdgcn_wmma_f32_16x16x4_f32(
                false, a[kk],     false, b[kk],     (short)0, acc0, false, false);
            acc1 = __builtin_amdgcn_wmma_f32_16x16x4_f32(
                false, a[kk + 1], false, b[kk + 1], (short)0, acc1, false, false);
        }

        #pragma unroll
        for (int v = 0; v < 8; ++v) {
            const float c  = (acc0[v] + acc1[v]) * rn[v];  // cosine
            const float c2 = c * c;
            const float k4 = c2 * c2 * cf;                 // cos^4 * alpha*d^2
            uacc[v] += (zv[v] == spec) ? k4 : 0.0f;        // species mask
        }

        // Rotate double buffer (register moves)
        #pragma unroll
        for (int kk = 0; kk < 16; ++kk) b[kk] = b2[kk];
        spec = spec2;
        cf   = cf2;
        m0   = m1;
    }

    // --- Reduce over the 16 columns held within each half-wave --------------
    #pragma unroll
    for (int v = 0; v < 8; ++v) {
        float s = uacc[v];
        s += __shfl_xor(s, 1, 32);
        s += __shfl_xor(s, 2, 32);
        s += __shfl_xor(s, 4, 32);
        s += __shfl_xor(s, 8, 32);
        uacc[v] = s;
    }

    // lane 0 -> rows 0..7 ; lane 16 -> rows 8..15 of this wave's partial
    if (l16 == 0) {
        #pragma unroll
        for (int v = 0; v < 8; ++v)
            red[wave][half * 8 + v] = uacc[v];
    }
    __syncthreads();

    // --- Combine the NWAVES partial U vectors, apply atom mask --------------
    if (tid < 16) {
        float s = 0.0f;
        #pragma unroll
        for (int w = 0; w < NWAVES; ++w)
            s += red[w][tid];
        const int row = atom0 + tid;
        U[row] = s * amask[row];
    }
}

// ---------------------------------------------------------------------------
extern "C" void kernel_launch(void* const* d_in, const int* in_sizes, int n_in,
                              void* d_out, int out_size, void* d_ws, size_t ws_size,
                              hipStream_t stream)
{
    const float* descriptors = (const float*)d_in[0];   // [16,512,64]
    const int*   Z_int       = (const int*)  d_in[1];   // [16,512]
    const float* atom_mask   = (const float*)d_in[2];   // [16,512]
    const float* sparse_q    = (const float*)d_in[3];   // [4000,64]
    const int*   sparse_spec = (const int*)  d_in[4];   // [4000]
    const float* alpha       = (const float*)d_in[5];   // [4000]
    const float* delta       = (const float*)d_in[6];   // [4]
    float*       U           = (float*)d_out;           // [16,512]

    // Workspace layout: rnorm [8192] | coef [4000]
    float* rnorm = (float*)d_ws;
    float* coef  = rnorm + NATOM;

    // Prep 1: inverse norms (8 waves per 256-thread block -> 8 atoms/block)
    tnep_rnorm_kernel<<<NATOM / 8, 256, 0, stream>>>(descriptors, rnorm);

    // Prep 2: coefficients
    tnep_coef_kernel<<<(NM + 255) / 256, 256, 0, stream>>>(
        sparse_spec, alpha, delta, coef);

    // Main: 512 blocks x 8 waves; block owns 16 atoms, waves split M
    tnep_u_kernel<<<NATOM / 16, 32 * NWAVES, 0, stream>>>(
        descriptors, Z_int, atom_mask, sparse_q, sparse_spec, rnorm, coef, U);
}